// MSGLayer_6682969112720
// MI455X (gfx1250) — compile-verified
//
#include <hip/hip_runtime.h>

// CDNA5 / gfx1250, wave32. All matrix math via V_WMMA_F32_16X16X4_F32 (f32-exact).
typedef __attribute__((ext_vector_type(2))) float v2f;
typedef __attribute__((ext_vector_type(8))) float v8f;

#define B_   32
#define N_   128
#define T_   4096
#define PN_  512
#define TP_  4112          // 257 * 16 rows of propagated series (incl. replication pad)
#define TPH_ 2056          // TP_/2 (row pairs)
#define NT1  257           // t-tiles per batch in prop kernel

// workspace layout (float offsets)
#define OFF_A1   0         // a1   = rownorm(adj + I)          128x128
#define OFF_A1S  16384     // a1^2
#define OFF_A2   32768     // a2   = rownorm(adj^T + I)
#define OFF_A2S  49152     // a2^2
#define OFF_U    65536     // U[5][16][16]: U0', U1, U2, V1, V2
#define OFF_BIAS 66816     // b1 + b2 (16)
#define OFF_Y    67072     // 4 streams of Y, pair-interleaved [b][t/2][v][2]
#define YSTRIDE  ((size_t)B_ * TP_ * N_)

// LDS pair-interleaved a-matrix: aP[w/2][v][2], pair-row stride 288 dwords.
// 288 % 64 == 32  ->  lower/upper half-wave b64 reads cover disjoint bank halves.
#define PAIRSTRIDE 288
#define PAIRMAT    (64 * PAIRSTRIDE)   // dwords per matrix (72 KB)
#define XROW       20                  // padded row for X staging (16B aligned, even)

#define ALPHA_ 0.05f
#define OMA_   0.95f

static __device__ __forceinline__
v8f wmma4(v2f a, v2f b, v8f c) {
  // D = A(16x4,f32) x B(4x16,f32) + C(16x16,f32)
  return __builtin_amdgcn_wmma_f32_16x16x4_f32(false, a, false, b, (short)0, c, false, false);
}

// ---------------------------------------------------------------------------
// Kernel 0: normalized adjacency (both orientations), their squares, folded
// projection matrices U and bias. One block; squares computed from LDS copies
// so no cross-thread global-visibility assumptions are needed.
// ---------------------------------------------------------------------------
__global__ void __launch_bounds__(256)
precompute_kernel(const float* __restrict__ adj,
                  const float* __restrict__ W1, const float* __restrict__ b1,
                  const float* __restrict__ W2, const float* __restrict__ b2,
                  float* __restrict__ ws) {
  extern __shared__ float sm[];               // 2 * 16384 floats (128 KB)
  float* a1L = sm;
  float* a2L = sm + 16384;
  __shared__ float rs1[128], rs2[128];

  const int tid = threadIdx.x;
  if (tid < 128) {
    float s1 = 1.0f, s2 = 1.0f;               // +1 for the identity diagonal
    for (int v = 0; v < 128; ++v) {
      s1 += adj[tid * 128 + v];               // row sum of adj
      s2 += adj[v * 128 + tid];               // row sum of adj^T (= col sum)
    }
    rs1[tid] = 1.0f / s1;
    rs2[tid] = 1.0f / s2;
  }
  __syncthreads();

  for (int idx = tid; idx < 16384; idx += 256) {
    int w = idx >> 7, v = idx & 127;
    float e = (w == v) ? 1.0f : 0.0f;
    float e1 = (adj[w * 128 + v] + e) * rs1[w];
    float e2 = (adj[v * 128 + w] + e) * rs2[w];
    a1L[idx] = e1;  ws[OFF_A1 + idx] = e1;
    a2L[idx] = e2;  ws[OFF_A2 + idx] = e2;
  }
  __syncthreads();

  for (int idx = tid; idx < 16384; idx += 256) {
    int w = idx >> 7, v = idx & 127;
    float s1 = 0.0f, s2 = 0.0f;
    for (int u = 0; u < 128; ++u) {
      s1 += a1L[w * 128 + u] * a1L[u * 128 + v];
      s2 += a2L[w * 128 + u] * a2L[u * 128 + v];
    }
    ws[OFF_A1S + idx] = s1;
    ws[OFF_A2S + idx] = s2;
  }

  // Folded projection matrices: out = U0'Z + U1(Za1) + U2(Za1^2) + V1(Za2) + V2(Za2^2)
  if (tid < 256) {
    int o = tid >> 4, l = tid & 15;
    float w1a = W1[o * 48 + l], w1b = W1[o * 48 + 16 + l], w1c = W1[o * 48 + 32 + l];
    float w2a = W2[o * 48 + l], w2b = W2[o * 48 + 16 + l], w2c = W2[o * 48 + 32 + l];
    float* U = ws + OFF_U;
    U[0 * 256 + tid] = w1a + ALPHA_ * (w1b + w1c) + w2a + ALPHA_ * (w2b + w2c);
    U[1 * 256 + tid] = OMA_ * (w1b + ALPHA_ * w1c);
    U[2 * 256 + tid] = OMA_ * OMA_ * w1c;
    U[3 * 256 + tid] = OMA_ * (w2b + ALPHA_ * w2c);
    U[4 * 256 + tid] = OMA_ * OMA_ * w2c;
  }
  if (tid < 16) ws[OFF_BIAS + tid] = b1[tid] + b2[tid];
}

// ---------------------------------------------------------------------------
// Kernel 1: node-dim propagation on the raw (replication-padded) time axis.
// Y_s[b][t][v] = sum_w x[b][w][min(t,4095)] * a_s[w][v]   for s in {a1,a1^2,a2,a2^2}
// Per wave: one 16-row t-tile, X^T held as 32 A-fragments (64 VGPRs, reused
// 32x across 4 streams x 8 v-tiles). B-fragments come from pair-interleaved
// LDS as single ds_load_b64 per WMMA; D-tiles stored as b64 pairs.
// 1024 v_wmma_f32_16x16x4_f32 per tile.
// ---------------------------------------------------------------------------
__global__ void __launch_bounds__(256)
prop_kernel(const float* __restrict__ x, float* __restrict__ ws) {
  extern __shared__ float aL[];               // 4 * PAIRMAT floats (288 KB)

  const int tid = threadIdx.x;
  for (int idx = tid; idx < 4 * 16384; idx += 256) {
    int s = idx >> 14;
    int r = idx & 16383;
    int w = r >> 7, v = r & 127;
    aL[s * PAIRMAT + (w >> 1) * PAIRSTRIDE + (v << 1) + (w & 1)] = ws[s * 16384 + r];
  }
  __syncthreads();

  const int wid  = tid >> 5;
  const int lane = tid & 31;
  const int half = lane >> 4;
  const int m16  = lane & 15;
  const int b    = blockIdx.x;
  const int mt   = blockIdx.y * 8 + wid;
  if (mt >= NT1) return;
  const int t0 = mt * 16;
  int tc = t0 + m16; if (tc > T_ - 1) tc = T_ - 1;   // replication pad

  // A-fragments of X^T (rows t, K = node w): chunk c covers w = 4c..4c+3.
  v2f af[32];
  const float* xb = x + (size_t)b * N_ * T_;
  #pragma unroll
  for (int c = 0; c < 32; ++c) {
    int w0 = 4 * c + 2 * half;
    af[c].x = xb[(size_t)w0 * T_ + tc];
    af[c].y = xb[(size_t)(w0 + 1) * T_ + tc];
  }

  float* Y = ws + OFF_Y;
  for (int s = 0; s < 4; ++s) {
    const float* am = aL + s * PAIRMAT;
    float* Ys = Y + (size_t)s * YSTRIDE;
    const size_t pbase = (size_t)b * TPH_ + (t0 >> 1) + 4 * half;
    for (int vt = 0; vt < 8; ++vt) {
      const int v = vt * 16 + m16;
      v8f acc0 = {}; v8f acc1 = {};              // 2 accumulators -> shorter dep chain
      #pragma unroll
      for (int c = 0; c < 32; c += 2) {
        // pair index q = w0/2 = 2c + half; one b64 = {a[w0][v], a[w0+1][v]}
        v2f bf0 = *(const v2f*)(am + (2 * c + half) * PAIRSTRIDE + 2 * v);
        acc0 = wmma4(af[c], bf0, acc0);
        v2f bf1 = *(const v2f*)(am + (2 * (c + 1) + half) * PAIRSTRIDE + 2 * v);
        acc1 = wmma4(af[c + 1], bf1, acc1);
      }
      // D layout: VGPR r -> row t0 + r + 8*half; rows (2rp, 2rp+1) -> one b64
      #pragma unroll
      for (int rp = 0; rp < 4; ++rp) {
        v2f val;
        val.x = acc0[2 * rp]     + acc1[2 * rp];
        val.y = acc0[2 * rp + 1] + acc1[2 * rp + 1];
        *(v2f*)(Ys + ((pbase + rp) * N_ + v) * 2) = val;
      }
    }
  }
}

// ---------------------------------------------------------------------------
// Kernel 2: per-patch projection. 4 waves/block, one patch per wave.
// out[b,o,p,n] = sum_l ( U0'[o,l] X[n,p8+l] + U1 Y1 + U2 Y1s + V1 Y2 + V2 Y2s ) + bias[o]
// X slice staged in LDS (float4 loads); Y B-fragments are single b64 loads
// from the pair-interleaved layout (half-wave = 128B contiguous).
// 160 v_wmma per wave-patch.
// ---------------------------------------------------------------------------
__global__ void __launch_bounds__(128)
proj_kernel(const float* __restrict__ x, const float* __restrict__ ws,
            float* __restrict__ out) {
  __shared__ float Xs[4 * 128 * XROW];          // 40 KB

  const int tid  = threadIdx.x;
  const int wid  = tid >> 5;
  const int lane = tid & 31;
  const int half = lane >> 4;
  const int m16  = lane & 15;
  const int b    = blockIdx.x;
  const int p    = blockIdx.y * 4 + wid;

  float* Xw = Xs + wid * 128 * XROW;            // Xw[n][l] = x[b][n][p*8+l] (clamped)
  const float* xb = x + (size_t)b * N_ * T_;
  if (p < PN_ - 1) {
    for (int i = 0; i < 4; ++i) {
      int n = i * 32 + lane;
      const float4* src = (const float4*)(xb + (size_t)n * T_ + p * 8);
      float4 q0 = src[0], q1 = src[1], q2 = src[2], q3 = src[3];
      float4* d = (float4*)(Xw + n * XROW);
      d[0] = q0; d[1] = q1; d[2] = q2; d[3] = q3;
    }
  } else {                                      // last patch: replication pad
    for (int i = 0; i < 4; ++i) {
      int n = i * 32 + lane;
      float* d = Xw + n * XROW;
      for (int l = 0; l < 16; ++l) {
        int t = p * 8 + l; if (t > T_ - 1) t = T_ - 1;
        d[l] = xb[(size_t)n * T_ + t];
      }
    }
  }
  __syncthreads();

  // Constant A-fragments for the five 16x16 projection matrices.
  const float* U = ws + OFF_U;
  v2f uf[5][4];
  #pragma unroll
  for (int s = 0; s < 5; ++s)
    #pragma unroll
    for (int c = 0; c < 4; ++c) {
      const float* up = U + (s * 16 + m16) * 16 + 4 * c + 2 * half;
      uf[s][c].x = up[0]; uf[s][c].y = up[1];
    }
  float bb[8];
  #pragma unroll
  for (int r = 0; r < 8; ++r) bb[r] = ws[OFF_BIAS + r + 8 * half];

  const float* Y = ws + OFF_Y;
  for (int nt = 0; nt < 8; ++nt) {
    const int n = nt * 16 + m16;
    v8f acc0 = {}; v8f acc1 = {};
    #pragma unroll
    for (int c = 0; c < 4; ++c) {
      const int l0 = 4 * c + 2 * half;            // even
      v2f bx = *(const v2f*)(Xw + n * XROW + l0); // {X[n][l0], X[n][l0+1]}
      acc0 = wmma4(uf[0][c], bx, acc0);
      // pair index tp = (p*8 + l0)/2 = p*4 + 2c + half
      const size_t tp = (size_t)p * 4 + 2 * c + half;
      const float* yb = Y + (((size_t)b * TPH_ + tp) * N_ + n) * 2;
      v2f y1 = *(const v2f*)(yb);                 // {Y[t][n], Y[t+1][n]}
      acc0 = wmma4(uf[1][c], y1, acc0);
      v2f y2 = *(const v2f*)(yb + YSTRIDE);
      acc0 = wmma4(uf[2][c], y2, acc0);
      v2f y3 = *(const v2f*)(yb + 2 * YSTRIDE);
      acc1 = wmma4(uf[3][c], y3, acc1);
      v2f y4 = *(const v2f*)(yb + 3 * YSTRIDE);
      acc1 = wmma4(uf[4][c], y4, acc1);
    }
    #pragma unroll
    for (int r = 0; r < 8; ++r) {         // D layout: o = r + 8*half, n = nt*16 + m16
      int o = r + 8 * half;
      out[(((size_t)b * 16 + o) * PN_ + p) * N_ + n] = acc0[r] + acc1[r] + bb[r];
    }
  }
}

// ---------------------------------------------------------------------------
extern "C" void kernel_launch(void* const* d_in, const int* in_sizes, int n_in,
                              void* d_out, int out_size, void* d_ws, size_t ws_size,
                              hipStream_t stream) {
  (void)in_sizes; (void)n_in; (void)out_size; (void)ws_size;
  const float* x   = (const float*)d_in[0];
  const float* adj = (const float*)d_in[1];
  const float* W1  = (const float*)d_in[2];
  const float* b1  = (const float*)d_in[3];
  const float* W2  = (const float*)d_in[4];
  const float* b2  = (const float*)d_in[5];
  float* out = (float*)d_out;
  float* ws  = (float*)d_ws;   // needs ~258 MiB: 4 Y streams + small tables

  (void)hipFuncSetAttribute((const void*)precompute_kernel,
                            hipFuncAttributeMaxDynamicSharedMemorySize,
                            2 * 16384 * (int)sizeof(float));
  (void)hipFuncSetAttribute((const void*)prop_kernel,
                            hipFuncAttributeMaxDynamicSharedMemorySize,
                            4 * PAIRMAT * (int)sizeof(float));

  precompute_kernel<<<1, 256, 2 * 16384 * sizeof(float), stream>>>(adj, W1, b1, W2, b2, ws);
  prop_kernel<<<dim3(B_, (NT1 + 7) / 8), 256, 4 * PAIRMAT * sizeof(float), stream>>>(x, ws);
  proj_kernel<<<dim3(B_, PN_ / 4), 128, 0, stream>>>(x, ws, out);
}